// ExtractLayer_7791070675546
// MI455X (gfx1250) — compile-verified
//
#include <hip/hip_runtime.h>

typedef __attribute__((ext_vector_type(2))) float v2f;
typedef __attribute__((ext_vector_type(8))) float v8f;

#define NEG_SLOPE 0.2f

__device__ __forceinline__ v8f wmma_f32(v2f a, v2f b, v8f c) {
    return __builtin_amdgcn_wmma_f32_16x16x4_f32(false, a, false, b, (short)0, c,
                                                 false, false);
}

// ---------------------------------------------------------------------------
// GEMM + bias: Y[N,C] = X[N,K] @ W[K,C] + b[C], fp32 WMMA 16x16x4.
// K,C are compile-time so all strides fold into immediate offsets.
// One wave computes a 32x32 output block (2x2 tiles, 4 accumulators):
// per K-step -> 2 contiguous b64 A loads + 4 b32 B loads feed 4 WMMAs.
// EXEC stays all-1s through every WMMA (wave-uniform guard; clamped loads,
// masked stores for the N%32 tail).
// ---------------------------------------------------------------------------
template <int K, int C>
__global__ void gemm_bias_wmma(const float* __restrict__ X,
                               const float* __restrict__ W,
                               const float* __restrict__ bias,
                               float* __restrict__ Y, int N) {
    const int tilesN32 = C / 32;
    const int tilesM32 = (N + 31) >> 5;
    const int waveId = blockIdx.x * (blockDim.x >> 5) + (threadIdx.x >> 5);
    if (waveId >= tilesM32 * tilesN32) return;      // wave-uniform
    const int tM   = waveId / tilesN32;
    const int tN   = waveId % tilesN32;
    const int lane = threadIdx.x & 31;
    const int half = lane >> 4;                     // 0: K {0,1}, 1: K {2,3}
    const int l16  = lane & 15;

    int r0 = tM * 32 + l16;                         // A rows, M-tile 0
    int r1 = r0 + 16;                               // A rows, M-tile 1
    if (r0 >= N) r0 = N - 1;                        // clamp tail (cndmask)
    if (r1 >= N) r1 = N - 1;
    const int c0 = tN * 32 + l16;                   // cols, N-tile 0 (C%32==0)
    const int c1 = c0 + 16;                         // cols, N-tile 1

    const float* __restrict__ x0 = X + (size_t)r0 * K;
    const float* __restrict__ x1 = X + (size_t)r1 * K;
    const float* __restrict__ w0 = W + c0;
    const float* __restrict__ w1 = W + c1;

    v8f acc00 = {0.f, 0.f, 0.f, 0.f, 0.f, 0.f, 0.f, 0.f};
    v8f acc01 = acc00, acc10 = acc00, acc11 = acc00;

#pragma unroll 4
    for (int k0 = 0; k0 < K; k0 += 4) {
        const int ka = k0 + (half << 1);
        // A 16x4 fp32 frag: lanes 0-15 K={k0,k0+1}, lanes 16-31 K={k0+2,k0+3};
        // the pair is contiguous in memory -> single b64 load.
        v2f a0 = *(const v2f*)(x0 + ka);
        v2f a1 = *(const v2f*)(x1 + ka);
        // B 4x16 fp32 frags (row stride C is a constant -> immediate offsets)
        v2f b0, b1;
        b0.x = w0[ka * C];
        b0.y = w0[(ka + 1) * C];
        b1.x = w1[ka * C];
        b1.y = w1[(ka + 1) * C];
        acc00 = wmma_f32(a0, b0, acc00);
        acc01 = wmma_f32(a0, b1, acc01);
        acc10 = wmma_f32(a1, b0, acc10);
        acc11 = wmma_f32(a1, b1, acc11);
    }

    const float bv0 = bias[c0];
    const float bv1 = bias[c1];
    // C/D layout: VGPR r -> M=r (lanes 0-15), M=8+r (lanes 16-31); N=l16
#pragma unroll
    for (int r = 0; r < 8; ++r) {
        const int m0 = tM * 32 + r + (half << 3);
        const int m1 = m0 + 16;
        if (m0 < N) {
            Y[(size_t)m0 * C + c0] = acc00[r] + bv0;
            Y[(size_t)m0 * C + c1] = acc01[r] + bv1;
        }
        if (m1 < N) {
            Y[(size_t)m1 * C + c0] = acc10[r] + bv0;
            Y[(size_t)m1 * C + c1] = acc11[r] + bv1;
        }
    }
}

// ---------------------------------------------------------------------------
// Output init: out = x + b1 (+ b2)   (residual + per-relation biases)
// ---------------------------------------------------------------------------
__global__ void init_out_kernel(const float* __restrict__ x,
                                const float* __restrict__ b1,
                                const float* __restrict__ b2,
                                float* __restrict__ out, int total, int C) {
    int i = blockIdx.x * blockDim.x + threadIdx.x;
    if (i >= total) return;
    int c = i % C;
    float v = x[i] + b1[c];
    if (b2) v += b2[c];
    out[i] = v;
}

// Zero segment-max (encoded unsigned) and denominator buffers.
__global__ void zero_seg_kernel(unsigned* __restrict__ emax_enc,
                                float* __restrict__ denom, int n) {
    int i = blockIdx.x * blockDim.x + threadIdx.x;
    if (i >= n) return;
    emax_enc[i] = 0u;       // smallest key in order-preserving encoding
    denom[i]    = 0.f;
}

__device__ __forceinline__ unsigned float_to_ordered(float f) {
    unsigned b = __float_as_uint(f);
    return (b & 0x80000000u) ? ~b : (b | 0x80000000u);
}
__device__ __forceinline__ float ordered_to_float(unsigned u) {
    unsigned b = (u & 0x80000000u) ? (u & 0x7FFFFFFFu) : ~u;
    return __uint_as_float(b);
}

// ---------------------------------------------------------------------------
// Pass 1: e[edge] = leakyrelu(xl[src]+xr[dst](+ef@We)) . att ; segment max
// One wave per edge; wave32 shuffle reduction; ordered-u32 atomic max.
// ---------------------------------------------------------------------------
__global__ void edge_logits_kernel(const float* __restrict__ xl,
                                   const float* __restrict__ xr,
                                   const int* __restrict__ src,
                                   const int* __restrict__ dst,
                                   const float* __restrict__ att,
                                   const float* __restrict__ ef,   // [E,8] or null
                                   const float* __restrict__ We,   // [8,C] or null
                                   float* __restrict__ e_out,
                                   unsigned* __restrict__ emax_enc,
                                   int E, int C) {
    const int edge = blockIdx.x * (blockDim.x >> 5) + (threadIdx.x >> 5);
    if (edge >= E) return;
    const int lane = threadIdx.x & 31;
    const int s = src[edge], d = dst[edge];
    const float* __restrict__ pl = xl + (size_t)s * C;
    const float* __restrict__ pr = xr + (size_t)d * C;

    const bool he = (ef != nullptr);
    float efr[8];
    if (he) {
#pragma unroll
        for (int j = 0; j < 8; ++j) efr[j] = ef[(size_t)edge * 8 + j];
    }

    float sum = 0.f;
    for (int c = lane; c < C; c += 32) {
        float m = pl[c] + pr[c];
        if (he) {
            float t = 0.f;
#pragma unroll
            for (int j = 0; j < 8; ++j) t += efr[j] * We[j * C + c];
            m += t;
        }
        float lr = m > 0.f ? m : m * NEG_SLOPE;
        sum += lr * att[c];
    }
#pragma unroll
    for (int off = 16; off > 0; off >>= 1) sum += __shfl_xor(sum, off, 32);

    if (lane == 0) {
        e_out[edge] = sum;
        atomicMax(emax_enc + d, float_to_ordered(sum));   // global_atomic_max_u32
    }
}

// ---------------------------------------------------------------------------
// Pass 2: ee = exp(e - emax[dst]); denom[dst] += ee
// ---------------------------------------------------------------------------
__global__ void edge_exp_kernel(const float* __restrict__ e_in,
                                const int* __restrict__ dst,
                                const unsigned* __restrict__ emax_enc,
                                float* __restrict__ ee_out,
                                float* __restrict__ denom, int E) {
    int i = blockIdx.x * blockDim.x + threadIdx.x;
    if (i >= E) return;
    const int d = dst[i];
    const float emax = ordered_to_float(emax_enc[d]);
    const float ee = __expf(e_in[i] - emax);
    ee_out[i] = ee;
    atomicAdd(denom + d, ee);                            // global_atomic_add_f32
}

// ---------------------------------------------------------------------------
// Pass 3: out[dst,:] += (ee/denom[dst]) * xl[src,:]   (one wave per edge)
// ---------------------------------------------------------------------------
__global__ void edge_aggregate_kernel(const float* __restrict__ ee,
                                      const float* __restrict__ denom,
                                      const float* __restrict__ xl,
                                      const int* __restrict__ src,
                                      const int* __restrict__ dst,
                                      float* __restrict__ out, int E, int C) {
    const int edge = blockIdx.x * (blockDim.x >> 5) + (threadIdx.x >> 5);
    if (edge >= E) return;
    const int lane = threadIdx.x & 31;
    const int s = src[edge], d = dst[edge];
    const float alpha = ee[edge] / fmaxf(denom[d], 1e-16f);
    const float* __restrict__ pl = xl + (size_t)s * C;
    float* __restrict__ po = out + (size_t)d * C;
    for (int c = lane; c < C; c += 32)
        atomicAdd(po + c, alpha * pl[c]);
}

// ---------------------------------------------------------------------------
// Host side
// ---------------------------------------------------------------------------
static void launch_gemm(hipStream_t stream, const float* X, const float* W,
                        const float* b, float* Y, int N, int K, int C) {
    const int tiles = ((N + 31) / 32) * (C / 32);
    const int blocks = (tiles + 7) / 8;            // 8 waves / 256-thread block
    if (K == 256 && C == 256)
        gemm_bias_wmma<256, 256><<<blocks, 256, 0, stream>>>(X, W, b, Y, N);
    else if (K == 256 && C == 128)
        gemm_bias_wmma<256, 128><<<blocks, 256, 0, stream>>>(X, W, b, Y, N);
    else
        gemm_bias_wmma<128, 128><<<blocks, 256, 0, stream>>>(X, W, b, Y, N);
}

static void run_gatv2(hipStream_t stream,
                      const float* x_src, int Ns, int Kin_src,
                      const float* x_dst, int Nd, int Kin_dst,
                      int C,
                      const float* Wl, const float* bl,
                      const float* Wr, const float* br,
                      const float* att,
                      const int* src, const int* dst, int E,
                      const float* ef, const float* We,
                      float* out,
                      float* xl, float* xr, float* eb,
                      unsigned* emx, float* den) {
    launch_gemm(stream, x_src, Wl, bl, xl, Ns, Kin_src, C);
    launch_gemm(stream, x_dst, Wr, br, xr, Nd, Kin_dst, C);
    zero_seg_kernel<<<(Nd + 255) / 256, 256, 0, stream>>>(emx, den, Nd);
    edge_logits_kernel<<<(E + 7) / 8, 256, 0, stream>>>(xl, xr, src, dst, att,
                                                        ef, We, eb, emx, E, C);
    edge_exp_kernel<<<(E + 255) / 256, 256, 0, stream>>>(eb, dst, emx, eb, den, E);
    edge_aggregate_kernel<<<(E + 7) / 8, 256, 0, stream>>>(eb, den, xl, src, dst,
                                                           out, E, C);
}

extern "C" void kernel_launch(void* const* d_in, const int* in_sizes, int n_in,
                              void* d_out, int out_size, void* d_ws, size_t ws_size,
                              hipStream_t stream) {
    // Problem constants (match reference setup_inputs)
    const int N_OP = 50000, N_M = 5000, N_A = 2000;
    const int C_OP = 256, C_M = 128, C_A = 128;
    const int E_OO = 200000, E_OM = 200000, E_MA = 60000;

    const float* x_op = (const float*)d_in[0];
    const float* x_m  = (const float*)d_in[1];
    const float* x_a  = (const float*)d_in[2];
    const int* ei_pred = (const int*)d_in[3];   // [2, E_OO]
    const int* ei_succ = (const int*)d_in[4];
    const int* src_o2m = (const int*)d_in[5];
    const int* dst_o2m = (const int*)d_in[6];
    const int* src_m2a = (const int*)d_in[7];
    const int* dst_m2a = (const int*)d_in[8];
    const float* ea_m2a = (const float*)d_in[9];

    const float* Wl_pred = (const float*)d_in[10];
    const float* bl_pred = (const float*)d_in[11];
    const float* Wr_pred = (const float*)d_in[12];
    const float* br_pred = (const float*)d_in[13];
    const float* att_pred = (const float*)d_in[14];
    const float* b_pred  = (const float*)d_in[15];

    const float* Wl_succ = (const float*)d_in[16];
    const float* bl_succ = (const float*)d_in[17];
    const float* Wr_succ = (const float*)d_in[18];
    const float* br_succ = (const float*)d_in[19];
    const float* att_succ = (const float*)d_in[20];
    const float* b_succ  = (const float*)d_in[21];

    const float* Wl_o2m = (const float*)d_in[22];
    const float* bl_o2m = (const float*)d_in[23];
    const float* Wr_o2m = (const float*)d_in[24];
    const float* br_o2m = (const float*)d_in[25];
    const float* att_o2m = (const float*)d_in[26];
    const float* b_o2m  = (const float*)d_in[27];

    const float* Wl_m2a = (const float*)d_in[28];
    const float* bl_m2a = (const float*)d_in[29];
    const float* Wr_m2a = (const float*)d_in[30];
    const float* br_m2a = (const float*)d_in[31];
    const float* att_m2a = (const float*)d_in[32];
    const float* b_m2a  = (const float*)d_in[33];
    const float* We_m2a = (const float*)d_in[34];

    float* out_op = (float*)d_out;                  // [50000,256]
    float* out_m  = out_op + (size_t)N_OP * C_OP;   // [5000,128]
    float* out_a  = out_m + (size_t)N_M * C_M;      // [2000,128]

    // Workspace layout (sequential reuse across relations): peak ~103.6 MB
    char* ws = (char*)d_ws;
    float*    xl  = (float*)(ws);                            // 50000*256 f32
    float*    xr  = (float*)(ws + 51200000);                 // 50000*256 f32
    float*    eb  = (float*)(ws + 102400000);                // 200000 f32
    unsigned* emx = (unsigned*)(ws + 103200000);             // 50000 u32
    float*    den = (float*)(ws + 103400000);                // 50000 f32

    // Residual + output biases
    {
        int t = N_OP * C_OP;
        init_out_kernel<<<(t + 255) / 256, 256, 0, stream>>>(x_op, b_pred, b_succ,
                                                             out_op, t, C_OP);
    }
    {
        int t = N_M * C_M;
        init_out_kernel<<<(t + 255) / 256, 256, 0, stream>>>(x_m, b_o2m, nullptr,
                                                             out_m, t, C_M);
    }
    {
        int t = N_A * C_A;
        init_out_kernel<<<(t + 255) / 256, 256, 0, stream>>>(x_a, b_m2a, nullptr,
                                                             out_a, t, C_A);
    }

    // op -> op (pred)
    run_gatv2(stream, x_op, N_OP, C_OP, x_op, N_OP, C_OP, C_OP,
              Wl_pred, bl_pred, Wr_pred, br_pred, att_pred,
              ei_pred, ei_pred + E_OO, E_OO, nullptr, nullptr,
              out_op, xl, xr, eb, emx, den);
    // op -> op (succ)
    run_gatv2(stream, x_op, N_OP, C_OP, x_op, N_OP, C_OP, C_OP,
              Wl_succ, bl_succ, Wr_succ, br_succ, att_succ,
              ei_succ, ei_succ + E_OO, E_OO, nullptr, nullptr,
              out_op, xl, xr, eb, emx, den);
    // op -> machine
    run_gatv2(stream, x_op, N_OP, C_OP, x_m, N_M, C_M, C_M,
              Wl_o2m, bl_o2m, Wr_o2m, br_o2m, att_o2m,
              src_o2m, dst_o2m, E_OM, nullptr, nullptr,
              out_m, xl, xr, eb, emx, den);
    // machine -> agv (with edge features)
    run_gatv2(stream, x_m, N_M, C_M, x_a, N_A, C_A, C_A,
              Wl_m2a, bl_m2a, Wr_m2a, br_m2a, att_m2a,
              src_m2a, dst_m2a, E_MA, ea_m2a, We_m2a,
              out_a, xl, xr, eb, emx, den);
}